// STGN_74363063763510
// MI455X (gfx1250) — compile-verified
//
#include <hip/hip_runtime.h>

#define T_DIM   32
#define N_NODES 10000
#define N_EDGES 160000
#define FDIM    64
#define ROWS    (T_DIM * N_NODES)     // 320000, divisible by 128

typedef float v2f __attribute__((ext_vector_type(2)));
typedef float v8f __attribute__((ext_vector_type(8)));

// ---------------- small utility kernels ----------------

__global__ void fill_zero_kernel(float* __restrict__ p, int n) {
    int i = blockIdx.x * blockDim.x + threadIdx.x;
    if (i < n) p[i] = 0.0f;
}

__global__ void deg_accum_kernel(const int* __restrict__ dst,
                                 const float* __restrict__ ew,
                                 float* __restrict__ deg) {
    int e = blockIdx.x * blockDim.x + threadIdx.x;
    if (e < N_EDGES) atomicAdd(&deg[dst[e]], ew[e]);
}

__global__ void norm_kernel(const int* __restrict__ src,
                            const int* __restrict__ dst,
                            const float* __restrict__ ew,
                            const float* __restrict__ deg,
                            float* __restrict__ norm) {
    int e = blockIdx.x * blockDim.x + threadIdx.x;
    if (e >= N_EDGES) return;
    float ds = deg[src[e]];
    float dd = deg[dst[e]];
    float is = ds > 0.0f ? rsqrtf(ds) : 0.0f;
    float id = dd > 0.0f ? rsqrtf(dd) : 0.0f;
    norm[e] = is * ew[e] * id;
}

// out[r, f] = bias[f] for all ROWS rows (pre-init before scatter-add)
__global__ void fill_bias_kernel(float* __restrict__ out,
                                 const float* __restrict__ bias) {
    int i = blockIdx.x * blockDim.x + threadIdx.x;
    if (i < ROWS * FDIM) out[i] = bias[i & (FDIM - 1)];
}

// ---------------- WMMA fp32 GEMM: OUT[ROWS,64] = (relu?)X[ROWS,64] @ W[64,64] --------
// Block = 256 threads = 8 waves; block computes a 128x64 tile; wave computes 16x64.
// Uses V_WMMA_F32_16X16X4_F32 (fp32 A/B, K=4), 16 K-steps x 4 N-tiles per wave.

#define LDSW 65   // padded stride (floats) to avoid 64-bank conflicts

template <bool RELU_IN>
__global__ __launch_bounds__(256) void gemm64_wmma_kernel(
        const float* __restrict__ X,
        const float* __restrict__ W,
        float* __restrict__ OUT) {
    __shared__ float Ws[64 * LDSW];    // W staged, row = k, col = n
    __shared__ float Xs[128 * LDSW];   // 128-row X tile

    const int tid  = threadIdx.x;
    const int lane = tid & 31;
    const int wave = tid >> 5;
    const int m    = lane & 15;    // row (A) / col (B,D) within 16
    const int kh   = lane >> 4;    // K half-select (0 or 1)

    const int rowBase = blockIdx.x * 128;

    // Stage W (coalesced)
    for (int i = tid; i < 64 * 64; i += 256)
        Ws[(i >> 6) * LDSW + (i & 63)] = W[i];

    // Stage X tile (coalesced contiguous copy), fusing ReLU for layer 2
    const float* Xblk = X + (size_t)rowBase * FDIM;
    for (int i = tid; i < 128 * 64; i += 256) {
        float v = Xblk[i];
        if (RELU_IN) v = fmaxf(v, 0.0f);
        Xs[(i >> 6) * LDSW + (i & 63)] = v;
    }
    __syncthreads();

    v8f acc0 = {}, acc1 = {}, acc2 = {}, acc3 = {};
    const int r0 = wave * 16;   // wave's row offset inside tile

#if __has_builtin(__builtin_amdgcn_wmma_f32_16x16x4_f32)
#pragma unroll
    for (int k = 0; k < 64; k += 4) {
        const int kk = k + 2 * kh;
        v2f a;
        a.x = Xs[(r0 + m) * LDSW + kk];
        a.y = Xs[(r0 + m) * LDSW + kk + 1];

        v2f b0, b1, b2, b3;
        b0.x = Ws[kk * LDSW +  0 + m]; b0.y = Ws[(kk + 1) * LDSW +  0 + m];
        b1.x = Ws[kk * LDSW + 16 + m]; b1.y = Ws[(kk + 1) * LDSW + 16 + m];
        b2.x = Ws[kk * LDSW + 32 + m]; b2.y = Ws[(kk + 1) * LDSW + 32 + m];
        b3.x = Ws[kk * LDSW + 48 + m]; b3.y = Ws[(kk + 1) * LDSW + 48 + m];

        acc0 = __builtin_amdgcn_wmma_f32_16x16x4_f32(false, a, false, b0, (short)0, acc0, false, false);
        acc1 = __builtin_amdgcn_wmma_f32_16x16x4_f32(false, a, false, b1, (short)0, acc1, false, false);
        acc2 = __builtin_amdgcn_wmma_f32_16x16x4_f32(false, a, false, b2, (short)0, acc2, false, false);
        acc3 = __builtin_amdgcn_wmma_f32_16x16x4_f32(false, a, false, b3, (short)0, acc3, false, false);
    }
#else
    // Scalar fallback (should not be taken on gfx1250; histogram would show wmma=0)
#pragma unroll 1
    for (int k = 0; k < 64; ++k) {
        for (int i = 0; i < 8; ++i) {
            float av = Xs[(r0 + i + 8 * kh) * LDSW + k];
            acc0[i] += av * Ws[k * LDSW +  0 + m];
            acc1[i] += av * Ws[k * LDSW + 16 + m];
            acc2[i] += av * Ws[k * LDSW + 32 + m];
            acc3[i] += av * Ws[k * LDSW + 48 + m];
        }
    }
#endif

    // D layout: VGPR i -> row (i + 8*kh), col = m (within each 16-col tile)
#pragma unroll
    for (int i = 0; i < 8; ++i) {
        const size_t row = (size_t)rowBase + r0 + i + 8 * kh;
        float* o = OUT + row * FDIM;
        o[ 0 + m] = acc0[i];
        o[16 + m] = acc1[i];
        o[32 + m] = acc2[i];
        o[48 + m] = acc3[i];
    }
}

// ---------------- edge scatter: out[t,dst,:] += h[t,src,:] * norm[e] ----------------
// 64 consecutive threads handle one (t,e): coalesced gather + coalesced atomics.

__global__ void scatter_kernel(const float* __restrict__ h,
                               const float* __restrict__ norm,
                               const int* __restrict__ src,
                               const int* __restrict__ dst,
                               float* __restrict__ out) {
    const long long total = (long long)T_DIM * N_EDGES * FDIM;
    long long idx = (long long)blockIdx.x * blockDim.x + threadIdx.x;
    if (idx >= total) return;
    const int f = (int)(idx & (FDIM - 1));
    const long long et = idx >> 6;
    const int e = (int)(et % N_EDGES);
    const int t = (int)(et / N_EDGES);
    const int s = src[e];
    const int d = dst[e];
    const float v = h[(t * N_NODES + s) * FDIM + f] * norm[e];
    atomicAdd(&out[(t * N_NODES + d) * FDIM + f], v);
}

// ---------------- launcher ----------------

extern "C" void kernel_launch(void* const* d_in, const int* in_sizes, int n_in,
                              void* d_out, int out_size, void* d_ws, size_t ws_size,
                              hipStream_t stream) {
    const float* x  = (const float*)d_in[0];
    const int*   ei = (const int*)d_in[1];       // (2, E) int32
    const float* ew = (const float*)d_in[2];
    const float* W1 = (const float*)d_in[3];
    const float* b1 = (const float*)d_in[4];
    const float* W2 = (const float*)d_in[5];
    const float* b2 = (const float*)d_in[6];
    float* out = (float*)d_out;

    const int* src = ei;
    const int* dst = ei + N_EDGES;

    float* ws   = (float*)d_ws;
    float* deg  = ws;                              // N_NODES
    float* norm = ws + N_NODES;                    // N_EDGES
    float* h    = norm + N_EDGES;                  // ROWS*FDIM
    float* agg  = h + (size_t)ROWS * FDIM;         // ROWS*FDIM

    const dim3 blk(256);
    const long long totScatter = (long long)T_DIM * N_EDGES * FDIM;
    const unsigned scatterBlocks = (unsigned)((totScatter + 255) / 256);

    // normalization
    fill_zero_kernel<<<(N_NODES + 255) / 256, blk, 0, stream>>>(deg, N_NODES);
    deg_accum_kernel<<<(N_EDGES + 255) / 256, blk, 0, stream>>>(dst, ew, deg);
    norm_kernel<<<(N_EDGES + 255) / 256, blk, 0, stream>>>(src, dst, ew, deg, norm);

    // layer 1: h = x @ W1 ; agg = b1 + scatter(h)  (ReLU folded into next GEMM's load)
    gemm64_wmma_kernel<false><<<ROWS / 128, blk, 0, stream>>>(x, W1, h);
    fill_bias_kernel<<<(ROWS * FDIM + 255) / 256, blk, 0, stream>>>(agg, b1);
    scatter_kernel<<<scatterBlocks, blk, 0, stream>>>(h, norm, src, dst, agg);

    // layer 2: h = relu(agg) @ W2 ; out = b2 + scatter(h)
    gemm64_wmma_kernel<true><<<ROWS / 128, blk, 0, stream>>>(agg, W2, h);
    fill_bias_kernel<<<(ROWS * FDIM + 255) / 256, blk, 0, stream>>>(out, b2);
    scatter_kernel<<<scatterBlocks, blk, 0, stream>>>(h, norm, src, dst, out);
}